// EdgePolicyModel_31971736552135
// MI455X (gfx1250) — compile-verified
//
#include <hip/hip_runtime.h>
#include <hip/hip_bf16.h>

typedef __attribute__((ext_vector_type(2))) float v2f;
typedef __attribute__((ext_vector_type(8))) float v8f;

// Native hardware f32 atomic add (global_atomic_add_f32, no CAS loop).
__device__ __forceinline__ void atom_add_f32(float* p, float v) {
  unsafeAtomicAdd(p, v);   // HIP maps this to the hardware global FP-add atomic
}

// ---------------- workspace zeroing (atomic accumulators + misc) -------------
__global__ void zero_kernel(unsigned int* __restrict__ p, long long n,
                            long long node_off) {
  long long i = (long long)blockIdx.x * blockDim.x + threadIdx.x;
  long long stride = (long long)gridDim.x * blockDim.x;
  for (; i < n; i += stride)
    p[i] = (i == node_off) ? 0x7fffffffu : 0u;   // node slot primed for atomicMin
}

// ---------------- out-degree over src (matches segment_sum over src) ---------
__global__ void degree_kernel(const int* __restrict__ src,
                              unsigned int* __restrict__ deg, long long E) {
  long long e = (long long)blockIdx.x * blockDim.x + threadIdx.x;
  if (e < E) {
    int s = __builtin_nontemporal_load(src + e);
    atomicAdd(&deg[s], 1u);
  }
}

// ---------------- find the marked node (first i with x[i,0]==1.0) -----------
__global__ void findnode_kernel(const float* __restrict__ x,
                                int* __restrict__ node_out, long long N) {
  long long i = (long long)blockIdx.x * blockDim.x + threadIdx.x;
  if (i < N && x[i * 4] == 1.0f) atomicMin(node_out, (int)i);
}

// ---------------- dinv = deg>0 ? rsqrt(deg) : 0 ------------------------------
__global__ void dinv_kernel(const unsigned int* __restrict__ deg,
                            float* __restrict__ dinv, long long N) {
  long long i = (long long)blockIdx.x * blockDim.x + threadIdx.x;
  if (i < N) {
    unsigned int d = deg[i];
    dinv[i] = d ? rsqrtf((float)d) : 0.0f;
  }
}

// ---------------- edge scatter: acc[dst] += -dinv[s]*dinv[d] * feat[src] -----
__global__ void scatter_kernel(const int* __restrict__ src,
                               const int* __restrict__ dst,
                               const float* __restrict__ dinv,
                               const float* __restrict__ feat,
                               float* __restrict__ acc, long long E) {
  long long e = (long long)blockIdx.x * blockDim.x + threadIdx.x;
  if (e >= E) return;
  int s = __builtin_nontemporal_load(src + e);   // streaming: keep L2 for hot data
  int d = __builtin_nontemporal_load(dst + e);
  float w = -dinv[s] * dinv[d];
  if (w != 0.0f) {
    const float4 xv = *reinterpret_cast<const float4*>(feat + (long long)s * 4);
    float* t = acc + (long long)d * 4;
    atom_add_f32(t + 0, w * xv.x);
    atom_add_f32(t + 1, w * xv.y);
    atom_add_f32(t + 2, w * xv.z);
    atom_add_f32(t + 3, w * xv.w);
  }
}

// ---------------- dense: h = relu(xin@W[0] + tx@W[1] + b) via WMMA f32 -------
// One wave handles a 16-node tile. A: 16x4 f32 (2 VGPR), B: 4x16 f32 (2 VGPR,
// W in cols 0..3, rest zero), C/D: 16x16 f32 (8 VGPR, bias preloaded).
__global__ void dense_wmma_kernel(const float* __restrict__ xin,
                                  const float* __restrict__ tx,
                                  const float* __restrict__ W,    // [2,4,4]
                                  const float* __restrict__ bias, // [4]
                                  float* __restrict__ hout, int nTiles) {
  int gwave = (int)(((long long)blockIdx.x * blockDim.x + threadIdx.x) >> 5);
  int lane  = threadIdx.x & 31;
  if (gwave >= nTiles) return;          // wave-uniform: EXEC stays all-1 inside

  int m    = lane & 15;                 // A row / B,C,D column index
  int half = lane >> 4;                 // K-half (A,B) / row-half (C,D)
  int kb   = half << 1;                 // K base: 0 or 2
  long long node = (long long)gwave * 16 + m;

  // A matrices: lane holds (K=kb, K=kb+1) of row m
  v2f a0, a1;
  const float* xr = xin + node * 4 + kb;
  a0.x = xr[0]; a0.y = xr[1];
  const float* tr = tx + node * 4 + kb;
  a1.x = tr[0]; a1.y = tr[1];

  // B matrices: lane holds (K=kb, K=kb+1) of column n; cols >=4 are zero
  int n = m;
  float b00 = 0.f, b01 = 0.f, b10 = 0.f, b11 = 0.f, bb = 0.f;
  if (n < 4) {
    b00 = W[kb * 4 + n];       b01 = W[(kb + 1) * 4 + n];        // W[0]
    b10 = W[16 + kb * 4 + n];  b11 = W[16 + (kb + 1) * 4 + n];   // W[1]
    bb  = bias[n];
  }
  v2f b0; b0.x = b00; b0.y = b01;
  v2f b1; b1.x = b10; b1.y = b11;

  v8f c;
#pragma unroll
  for (int v = 0; v < 8; ++v) c[v] = bb;   // bias broadcast along rows

  // D = tx@W1 + (x@W0 + C)
  c = __builtin_amdgcn_wmma_f32_16x16x4_f32(false, a0, false, b0, (short)0, c,
                                            false, false);
  c = __builtin_amdgcn_wmma_f32_16x16x4_f32(false, a1, false, b1, (short)0, c,
                                            false, false);

  if (n < 4) {
#pragma unroll
    for (int v = 0; v < 8; ++v) {
      float val = c[v];
      val = val > 0.f ? val : 0.f;         // relu
      hout[((long long)gwave * 16 + v + half * 8) * 4 + n] = val;
    }
  }
}

// ---------------- collect edges incident to the marked node ------------------
__global__ void collect_kernel(const int* __restrict__ src,
                               const int* __restrict__ dst,
                               const int* __restrict__ node_ptr,
                               unsigned int* __restrict__ counter,
                               int* __restrict__ slots, long long E) {
  long long e = (long long)blockIdx.x * blockDim.x + threadIdx.x;
  if (e >= E) return;
  int node = *node_ptr;
  int s = __builtin_nontemporal_load(src + e);
  int d = __builtin_nontemporal_load(dst + e);
  if (s == node || d == node) {
    unsigned int slot = atomicAdd(counter, 1u);
    if (slot < 64u) slots[slot] = (int)e;
  }
}

// ---------------- readout: rank-sort 64 edges, score, softmax ----------------
__global__ void readout_kernel(const int* __restrict__ src,
                               const int* __restrict__ dst,
                               const float* __restrict__ h2,
                               const float* __restrict__ Wr,
                               const float* __restrict__ br,
                               const int* __restrict__ node_ptr,
                               const int* __restrict__ slots,
                               float* __restrict__ out) {
  __shared__ float s_score[64];
  __shared__ int   s_tgt[64];
  __shared__ float red[64];
  int t = threadIdx.x;                 // 64 threads
  int node = *node_ptr;
  int e = slots[t];
  // rank among collected edges = index order of jnp.nonzero (edge ids unique)
  int rank = 0;
  for (int j = 0; j < 64; ++j) rank += (slots[j] < e) ? 1 : 0;
  int s = src[e], d = dst[e];
  int tgt = (s == node) ? d : s;
  const float* hv = h2 + (long long)tgt * 4;
  float sc = hv[0] * Wr[0] + hv[1] * Wr[1] + hv[2] * Wr[2] + hv[3] * Wr[3] + br[0];
  s_score[rank] = sc;
  s_tgt[rank]   = tgt;
  __syncthreads();

  float v = s_score[t];
  red[t] = v; __syncthreads();
  for (int off = 32; off > 0; off >>= 1) {
    if (t < off) red[t] = fmaxf(red[t], red[t + off]);
    __syncthreads();
  }
  float mx = red[0]; __syncthreads();
  float ex = expf(v - mx);
  red[t] = ex; __syncthreads();
  for (int off = 32; off > 0; off >>= 1) {
    if (t < off) red[t] += red[t + off];
    __syncthreads();
  }
  float sum = red[0];
  out[t]      = ex / sum;            // score
  out[64 + t] = (float)s_tgt[t];     // targets (as float in d_out)
}

// ---------------- driver -----------------------------------------------------
extern "C" void kernel_launch(void* const* d_in, const int* in_sizes, int n_in,
                              void* d_out, int out_size, void* d_ws, size_t ws_size,
                              hipStream_t stream) {
  const float* x  = (const float*)d_in[0];
  const int*   ei = (const int*)d_in[1];
  const float* W1 = (const float*)d_in[2];
  const float* b1 = (const float*)d_in[3];
  const float* W2 = (const float*)d_in[4];
  const float* b2 = (const float*)d_in[5];
  const float* Wr = (const float*)d_in[6];
  const float* br = (const float*)d_in[7];
  float* out = (float*)d_out;

  const long long N = in_sizes[0] / 4;
  const long long E = in_sizes[1] / 2;
  const int* src = ei;
  const int* dst = ei + E;

  // workspace layout (dwords): [deg N][tx 4N][tx2 4N][misc 128][dinv N][h1 4N][h2 4N]
  unsigned int* ws = (unsigned int*)d_ws;
  long long off = 0;
  unsigned int* deg = ws + off;          off += N;
  float* tx   = (float*)(ws + off);      off += 4 * N;
  float* tx2  = (float*)(ws + off);      off += 4 * N;
  unsigned int* misc = ws + off;         off += 128;
  const long long zeroCount = off;       // deg + tx + tx2 + misc
  float* dinv = (float*)(ws + off);      off += N;
  float* h1   = (float*)(ws + off);      off += 4 * N;
  float* h2   = (float*)(ws + off);      off += 4 * N;

  unsigned int* counter = misc + 0;
  int* node_idx = (int*)(misc + 1);
  int* slots    = (int*)(misc + 16);
  const long long node_off = (long long)((unsigned int*)node_idx - ws);

  const int blk = 256;
  const int gridE = (int)((E + blk - 1) / blk);
  const int gridN = (int)((N + blk - 1) / blk);
  const int nTiles = (int)((N + 15) / 16);
  const int gridD = (nTiles + 7) / 8;    // 8 waves (tiles) per 256-thread block

  zero_kernel<<<2048, blk, 0, stream>>>(ws, zeroCount, node_off);
  degree_kernel<<<gridE, blk, 0, stream>>>(src, deg, E);
  findnode_kernel<<<gridN, blk, 0, stream>>>(x, node_idx, N);
  dinv_kernel<<<gridN, blk, 0, stream>>>(deg, dinv, N);

  // layer 1
  scatter_kernel<<<gridE, blk, 0, stream>>>(src, dst, dinv, x, tx, E);
  dense_wmma_kernel<<<gridD, blk, 0, stream>>>(x, tx, W1, b1, h1, nTiles);
  // layer 2
  scatter_kernel<<<gridE, blk, 0, stream>>>(src, dst, dinv, h1, tx2, E);
  dense_wmma_kernel<<<gridD, blk, 0, stream>>>(h1, tx2, W2, b2, h2, nTiles);

  // readout
  collect_kernel<<<gridE, blk, 0, stream>>>(src, dst, node_idx, counter, slots, E);
  readout_kernel<<<1, 64, 0, stream>>>(src, dst, h2, Wr, br, node_idx, slots, out);
}